// KernelAttention_26079041421807
// MI455X (gfx1250) — compile-verified
//
#include <hip/hip_runtime.h>
#include <math.h>

typedef float v2f __attribute__((ext_vector_type(2)));
typedef float v8f __attribute__((ext_vector_type(8)));

#define B_   4
#define C_   256
#define H_   64
#define W_   64
#define HW_  4096
#define CI_  128
#define P_   9
#define NPIX (B_ * HW_)   // 16384

// ---------------------------------------------------------------------------
// wv_eff[c] = sum_o Wo[o] * Wv[o,c];  wobv = sum_o Wo[o] * bv[o]
// ---------------------------------------------------------------------------
__global__ void kv_wveff(const float* __restrict__ Wv, const float* __restrict__ Wo,
                         const float* __restrict__ bv, float* __restrict__ wv_eff,
                         float* __restrict__ wobv) {
  int c = threadIdx.x;                       // 256 threads
  float s = 0.f;
  for (int o = 0; o < CI_; ++o) s += Wo[o] * Wv[o * C_ + c];
  wv_eff[c] = s;
  if (c == 0) {
    float t = 0.f;
    for (int o = 0; o < CI_; ++o) t += Wo[o] * bv[o];
    *wobv = t;
  }
}

// ---------------------------------------------------------------------------
// vproj[b,hw] = sum_c wv_eff[c] * x[b,c,hw] + wobv     (== Wo . v[b,:,hw])
// ---------------------------------------------------------------------------
__global__ void kv_vproj(const float* __restrict__ x, const float* __restrict__ wv_eff,
                         const float* __restrict__ wobv, float* __restrict__ vproj) {
  int p = blockIdx.x * blockDim.x + threadIdx.x;
  if (p >= NPIX) return;
  int b = p >> 12, hw = p & (HW_ - 1);
  const float* xb = x + (size_t)b * C_ * HW_ + hw;
  float s = *wobv;
#pragma unroll 8
  for (int c = 0; c < C_; ++c) s += wv_eff[c] * xb[(size_t)c * HW_];
  vproj[p] = s;
}

// ---------------------------------------------------------------------------
// Bilinear tap metadata: 4 corner indices (clamped, 0 if invalid) + 4 weights
// (zeroed if corner outside the image) per (pixel, tap).
// ---------------------------------------------------------------------------
__global__ void k_taps(const float* __restrict__ offset, int4* __restrict__ tidx,
                       float4* __restrict__ twt) {
  int g = blockIdx.x * blockDim.x + threadIdx.x;      // g = pixel*9 + tap
  if (g >= NPIX * P_) return;
  int t = g % P_;
  int p = g / P_;
  int b = p >> 12, hw = p & (HW_ - 1);
  int h = hw >> 6, w = hw & 63;
  float oy = offset[((size_t)b * 18 + 2 * t) * HW_ + hw];
  float ox = offset[((size_t)b * 18 + 2 * t + 1) * HW_ + hw];
  float py = (float)h + (float)(t / 3 - 1) + oy;
  float px = (float)w + (float)(t % 3 - 1) + ox;
  float y0f = floorf(py), x0f = floorf(px);
  float fy = py - y0f, fx = px - x0f;
  int y0 = (int)y0f, x0 = (int)x0f;
  int y1 = y0 + 1, x1 = x0 + 1;
  float w00 = (1.f - fy) * (1.f - fx), w01 = (1.f - fy) * fx;
  float w10 = fy * (1.f - fx),         w11 = fy * fx;
  bool vy0 = (y0 >= 0) && (y0 < H_), vy1 = (y1 >= 0) && (y1 < H_);
  bool vx0 = (x0 >= 0) && (x0 < W_), vx1 = (x1 >= 0) && (x1 < W_);
  int4 I; float4 Wt;
  I.x = (vy0 && vx0) ? (y0 * W_ + x0) : 0;  Wt.x = (vy0 && vx0) ? w00 : 0.f;
  I.y = (vy0 && vx1) ? (y0 * W_ + x1) : 0;  Wt.y = (vy0 && vx1) ? w01 : 0.f;
  I.z = (vy1 && vx0) ? (y1 * W_ + x0) : 0;  Wt.z = (vy1 && vx0) ? w10 : 0.f;
  I.w = (vy1 && vx1) ? (y1 * W_ + x1) : 0;  Wt.w = (vy1 && vx1) ? w11 : 0.f;
  tidx[g] = I; twt[g] = Wt;
}

// ---------------------------------------------------------------------------
// q/k projection via V_WMMA_F32_16X16X4_F32.
// out layout transposed: out[b][hw][o]  (channel-contiguous for attention)
// Block = 256 threads = 8 waves; each wave owns one 16-row (o) tile; the
// 16-column (pixel) X tile is staged in LDS once per block.
// ---------------------------------------------------------------------------
__global__ __launch_bounds__(256) void k_proj(const float* __restrict__ x,
    const float* __restrict__ Wq, const float* __restrict__ bq,
    const float* __restrict__ Wk, const float* __restrict__ bk,
    float* __restrict__ qT, float* __restrict__ kT) {
  const float* Wm; const float* bias; float* out;
  if (blockIdx.y == 0) { Wm = Wq; bias = bq; out = qT; }
  else                 { Wm = Wk; bias = bk; out = kT; }

  __shared__ float xs[C_ * 16];                       // [c][n], 16 KB
  int tid = threadIdx.x;
  int n0  = blockIdx.x * 16;                          // global pixel column
  int b   = n0 >> 12;
  int hw0 = n0 & (HW_ - 1);
  const float* xb = x + (size_t)b * C_ * HW_ + hw0;
  {
    int nn = tid & 15, cb = tid >> 4;
    for (int r = 0; r < 16; ++r) {
      int c = cb + (r << 4);
      xs[c * 16 + nn] = xb[(size_t)c * HW_ + nn];     // coalesced per half-wave
    }
  }
  __syncthreads();

  int lane = tid & 31, wave = tid >> 5;
  int hi = lane >> 4, ln = lane & 15;
  int m0 = wave * 16;                                 // 8 waves cover o=0..127
  const float* wrow = Wm + (size_t)(m0 + ln) * C_ + 2 * hi;
  v8f acc = {};
#pragma unroll 8
  for (int k0 = 0; k0 < C_; k0 += 4) {
    v2f a = *(const v2f*)(wrow + k0);                 // A: lane=M row, K pair
    int cb0 = k0 + 2 * hi;                            // B: lane=N col, K pair
    v2f bb;
    bb.x = xs[cb0 * 16 + ln];
    bb.y = xs[(cb0 + 1) * 16 + ln];
    acc = __builtin_amdgcn_wmma_f32_16x16x4_f32(false, a, false, bb,
                                                (short)0, acc, false, false);
  }

  int col = n0 + ln;                                  // pixel this lane holds
  int hwc = col & (HW_ - 1);
  float* orow = out + ((size_t)(col >> 12) * HW_ + hwc) * CI_;
#pragma unroll
  for (int g = 0; g < 8; ++g) {
    int o = m0 + g + 8 * hi;                          // C layout: row g / g+8
    orow[o] = acc[g] + bias[o];
  }
}

// ---------------------------------------------------------------------------
// Fused deformable 9x9 attention. One wave per pixel.
// scores = qu^T ku accumulated with 32 x V_WMMA_F32_16X16X4_F32
// (A: M=tap i (9->16 pad), K=channel; B: K=channel, N=tap j).
// Then per-row softmax via xor-butterfly (stays inside 16-lane halves) fused
// with the Sum_j attn * s_v reduction; sigmoid; one store per row.
// ---------------------------------------------------------------------------
__global__ __launch_bounds__(128) void k_attn(const float* __restrict__ qT,
    const float* __restrict__ kT, const float* __restrict__ vproj,
    const int4* __restrict__ tidx, const float4* __restrict__ twt,
    const float* __restrict__ bo, float* __restrict__ out) {
  int lane = threadIdx.x & 31;
  int wave = threadIdx.x >> 5;
  int p = blockIdx.x * 4 + wave;                      // grid sized exactly
  int b = p >> 12, hw = p & (HW_ - 1);
  int t  = lane & 15;                                 // this lane's tap
  int hi = lane >> 4;

  int4   I  = make_int4(0, 0, 0, 0);
  float4 Wt = make_float4(0.f, 0.f, 0.f, 0.f);
  if (t < P_) { I = tidx[p * P_ + t]; Wt = twt[p * P_ + t]; }

  const float* qb = qT + (size_t)b * HW_ * CI_;
  const float* kb = kT + (size_t)b * HW_ * CI_;
  size_t a0 = (size_t)I.x * CI_, a1 = (size_t)I.y * CI_;
  size_t a2 = (size_t)I.z * CI_, a3 = (size_t)I.w * CI_;

  v8f acc = {};
#pragma unroll 4
  for (int k0 = 0; k0 < CI_; k0 += 4) {
    int c = k0 + 2 * hi;                              // channel pair this lane
    v2f qa = Wt.x * *(const v2f*)(qb + a0 + c)
           + Wt.y * *(const v2f*)(qb + a1 + c)
           + Wt.z * *(const v2f*)(qb + a2 + c)
           + Wt.w * *(const v2f*)(qb + a3 + c);       // qu[c..c+1, t]
    v2f ka = Wt.x * *(const v2f*)(kb + a0 + c)
           + Wt.y * *(const v2f*)(kb + a1 + c)
           + Wt.z * *(const v2f*)(kb + a2 + c)
           + Wt.w * *(const v2f*)(kb + a3 + c);       // ku[c..c+1, t]
    acc = __builtin_amdgcn_wmma_f32_16x16x4_f32(false, qa, false, ka,
                                                (short)0, acc, false, false);
  }

  // s_v[j] = Wo . vu[:, j]  == bilinear sample of vproj at tap j (= this lane)
  const float* vb = vproj + (size_t)b * HW_;
  float sv = Wt.x * vb[I.x] + Wt.y * vb[I.y] + Wt.z * vb[I.z] + Wt.w * vb[I.w];
  float bo0 = bo[0];
  float* ob = out + (size_t)b * P_ * HW_ + hw;
  const float NEG_INF = -__builtin_inff();

#pragma unroll
  for (int g = 0; g < 8; ++g) {
    int  i    = g + 8 * hi;                           // C layout: row g / g+8
    bool rowv = (i < P_);
    float s = acc[g];
    float m = (t < P_ && rowv) ? s : NEG_INF;
    for (int off = 8; off; off >>= 1)                 // max over 9 cols
      m = fmaxf(m, __shfl_xor(m, off, 32));
    float e   = (t < P_ && rowv) ? __expf(s - m) : 0.f;
    float num = e * sv;
    float den = e;
    for (int off = 8; off; off >>= 1) {               // fused sums
      den += __shfl_xor(den, off, 32);
      num += __shfl_xor(num, off, 32);
    }
    if (rowv && t == 0) {
      float val = num / den + bo0;
      ob[(size_t)i * HW_] = 1.f / (1.f + __expf(-val));
    }
  }
}

// ---------------------------------------------------------------------------
extern "C" void kernel_launch(void* const* d_in, const int* in_sizes, int n_in,
                              void* d_out, int out_size, void* d_ws, size_t ws_size,
                              hipStream_t stream) {
  const float* x      = (const float*)d_in[0];
  const float* offset = (const float*)d_in[1];
  const float* Wq     = (const float*)d_in[2];
  const float* bq     = (const float*)d_in[3];
  const float* Wk     = (const float*)d_in[4];
  const float* bk     = (const float*)d_in[5];
  const float* Wv     = (const float*)d_in[6];
  const float* bv     = (const float*)d_in[7];
  const float* Wo     = (const float*)d_in[8];
  const float* bo     = (const float*)d_in[9];
  float* out = (float*)d_out;

  char* ws = (char*)d_ws;
  size_t off = 0;
  float*  qT     = (float*)(ws + off); off += (size_t)NPIX * CI_ * sizeof(float); // 8 MB
  float*  kT     = (float*)(ws + off); off += (size_t)NPIX * CI_ * sizeof(float); // 8 MB
  int4*   tidx   = (int4*)(ws + off);  off += (size_t)NPIX * P_ * sizeof(int4);
  float4* twt    = (float4*)(ws + off);off += (size_t)NPIX * P_ * sizeof(float4);
  float*  vproj  = (float*)(ws + off); off += (size_t)NPIX * sizeof(float);
  float*  wv_eff = (float*)(ws + off); off += C_ * sizeof(float);
  float*  wobv   = (float*)(ws + off); off += 16;

  kv_wveff<<<1, 256, 0, stream>>>(Wv, Wo, bv, wv_eff, wobv);
  kv_vproj<<<NPIX / 256, 256, 0, stream>>>(x, wv_eff, wobv, vproj);
  k_taps<<<(NPIX * P_ + 255) / 256, 256, 0, stream>>>(offset, tidx, twt);
  k_proj<<<dim3(NPIX / 16, 2), 256, 0, stream>>>(x, Wq, bq, Wk, bk, qT, kT);
  k_attn<<<NPIX / 4, 128, 0, stream>>>(qT, kT, vproj, tidx, twt, bo, out);
}